// Gate_Attention_68126771249265
// MI455X (gfx1250) — compile-verified
//
#include <hip/hip_runtime.h>

// Problem constants
#define E_   158
#define F_   2049
#define T_   64
#define B_   8
#define EF_  (E_ * F_)        // 323742
#define EFT_ (EF_ * T_)       // 20719488

typedef float v2f __attribute__((ext_vector_type(2)));
typedef float v4f __attribute__((ext_vector_type(4)));
typedef float v8f __attribute__((ext_vector_type(8)));

// ---------------------------------------------------------------------------
// Kernel 1: proj[b,k] = x0*W[k,0] + x1*W[k,1] + bias[k]  -> anchor output
//           also zero Rx[0..255] (block 0's 256 threads) of the 512-float Rx accumulator.
// ---------------------------------------------------------------------------
__global__ void proj_kernel(const float* __restrict__ xind,
                            const float* __restrict__ W,
                            const float* __restrict__ bias,
                            float* __restrict__ anchor,
                            float* __restrict__ Rx) {
  if (blockIdx.x == 0 && threadIdx.x < B_ * T_) Rx[threadIdx.x] = 0.0f;
  int idx = blockIdx.x * blockDim.x + threadIdx.x;
  if (idx >= B_ * EF_) return;
  int b = idx / EF_;
  int k = idx - b * EF_;
  float x0 = xind[b * 2 + 0];
  float x1 = xind[b * 2 + 1];
  v2f w = *(const v2f*)(W + k * 2);                // b64 load of W[k,0..1]
  anchor[idx] = fmaf(x0, w.x, fmaf(x1, w.y, bias[k]));
}

// ---------------------------------------------------------------------------
// Kernel 2: Rx[b,t] = sum_k xs[b][t*EF + k] * proj[b][k] via WMMA f32 16x16x4.
//  A tile  : 16 t-rows x 4 k   (lane L<16: row=t0+L, K=0,1 ; L>=16: row=t0+L-16, K=2,3)
//  B tile  : proj broadcast over all 16 N columns
//  D tile  : every column n holds the same partial Rx for its 16 rows.
// Loop bounds are made wave-uniform SGPRs via readfirstlane (scalar branch),
// and the body is software-pipelined so next-iteration loads overlap the wmma.
// The single partial k-chunk (EF % 4 == 2) is peeled into one masked tail.
// ---------------------------------------------------------------------------
#define NSPLIT     64
#define NC4        ((EF_ + 3) / 4)                  // 80936 k-chunks of 4 (last holds 2)
#define CHUNKS_PER ((NC4 + NSPLIT - 1) / NSPLIT)    // 1265

__global__ void rx_wmma_kernel(const float* __restrict__ xs,
                               const float* __restrict__ proj,
                               float* __restrict__ Rx) {
  const int lane = threadIdx.x & 31;
  // wave index is uniform across the wave: force it into an SGPR so the k-loop
  // bounds are scalar (s_cbranch loop instead of EXEC-masked vector loop).
  const int wave = __builtin_amdgcn_readfirstlane((int)(threadIdx.x >> 5));
  const int bid  = blockIdx.x;                     // B_ * 4 ttiles * 8 splitblocks = 256
  const int b     = bid >> 5;
  const int rem   = bid & 31;
  const int ttile = rem >> 3;
  const int sblk  = rem & 7;
  const int split = sblk * 8 + wave;               // 0..63 (uniform)

  const int row  = ttile * 16 + (lane & 15);       // t index this lane feeds
  const int koff = (lane < 16) ? 0 : 2;            // K sub-offset within chunk
  const float* __restrict__ rp = xs   + (size_t)b * EFT_ + (size_t)row * EF_;
  const float* __restrict__ pp = proj + (size_t)b * EF_;
  const float* __restrict__ ap = rp + koff;
  const float* __restrict__ bp = pp + koff;

  const int cstart = split * CHUNKS_PER;
  int cend  = cstart + CHUNKS_PER;
  if (cend > NC4) cend = NC4;
  const int cfull = (cend < NC4 - 1) ? cend : (NC4 - 1); // all-full chunks only

  v8f acc = {};
  int c = cstart;
  if (c < cfull) {
    v2f a  = *(const v2f*)(ap + 4 * c);
    v2f bb = *(const v2f*)(bp + 4 * c);
    for (++c; c < cfull; ++c) {
      v2f an = *(const v2f*)(ap + 4 * c);          // prefetch next chunk
      v2f bn = *(const v2f*)(bp + 4 * c);
      acc = __builtin_amdgcn_wmma_f32_16x16x4_f32(false, a, false, bb,
                                                  (short)0, acc, false, false);
      a = an; bb = bn;
    }
    acc = __builtin_amdgcn_wmma_f32_16x16x4_f32(false, a, false, bb,
                                                (short)0, acc, false, false);
  }

  // Peeled tail: only the wave owning the last chunk gets here. kb..kb+1 are
  // valid (lanes < 16, koff==0); kb+2..kb+3 are OOB -> those lanes feed zeros.
  if (cend == NC4) {
    const int kb = (NC4 - 1) * 4;                  // 323740
    v2f a  = {0.0f, 0.0f};
    v2f bb = {0.0f, 0.0f};
    if (lane < 16) {
      a  = *(const v2f*)(rp + kb);
      bb = *(const v2f*)(pp + kb);
    }
    acc = __builtin_amdgcn_wmma_f32_16x16x4_f32(false, a, false, bb,
                                                (short)0, acc, false, false);
  }

  // D layout: lanes 0-15 VGPR r -> M=r ; lanes 16-31 VGPR r -> M=8+r. Columns identical.
  if ((lane & 15) == 0) {
    const int rbase = ttile * 16 + ((lane >= 16) ? 8 : 0);
    float* rxp = Rx + b * T_ + rbase;
#pragma unroll
    for (int r = 0; r < 8; ++r) atomicAdd(&rxp[r], acc[r]);
  }
}

// ---------------------------------------------------------------------------
// Kernel 3: out_flat[b][i] = xs_flat[b][i] * proj[b][i/64] * Rx[b][i%64]
// float4 (b128) loads, nontemporal b128 stores for the streamed output.
// ---------------------------------------------------------------------------
__global__ void gate_kernel(const float* __restrict__ xs,
                            const float* __restrict__ proj,
                            const float* __restrict__ Rx,
                            float* __restrict__ out) {
  long vid = (long)blockIdx.x * blockDim.x + threadIdx.x;
  long i = vid * 4;
  if (i >= (long)B_ * EFT_) return;
  int b = (int)(i / EFT_);
  int r = (int)(i - (long)b * EFT_);
  int k = r >> 6;                                  // / T_
  int t = r & 63;                                  // % T_ (multiple of 4)
  v4f x  = *(const v4f*)(xs + i);
  v4f rx = *(const v4f*)(Rx + b * T_ + t);
  float p = proj[(size_t)b * EF_ + k];
  v4f o = x * (rx * p);
  __builtin_nontemporal_store(o, (v4f*)(out + i));
}

// ---------------------------------------------------------------------------
extern "C" void kernel_launch(void* const* d_in, const int* in_sizes, int n_in,
                              void* d_out, int out_size, void* d_ws, size_t ws_size,
                              hipStream_t stream) {
  const float* xs   = (const float*)d_in[0];   // [B, E, F, T]
  const float* xind = (const float*)d_in[1];   // [B, C]
  const float* W    = (const float*)d_in[2];   // [E*F, C]
  const float* bias = (const float*)d_in[3];   // [E*F]

  float* out    = (float*)d_out;                       // B*E*F*T floats
  float* anchor = out + (size_t)B_ * EFT_;             // B*E*F floats (== proj)
  float* Rx     = (float*)d_ws;                        // B*T floats scratch

  // 1) proj -> anchor, zero Rx
  proj_kernel<<<(B_ * EF_ + 255) / 256, 256, 0, stream>>>(xind, W, bias, anchor, Rx);

  // 2) Rx reduction via WMMA (256 blocks x 8 waves = 2048 K-split waves)
  rx_wmma_kernel<<<B_ * 4 * 8, 256, 0, stream>>>(xs, anchor, Rx);

  // 3) gating pass (exactly B*EFT/4 vec4 elements; divides evenly by 256)
  const int nvec = B_ * EFT_ / 4;                      // 41,438,976
  gate_kernel<<<(nvec + 255) / 256, 256, 0, stream>>>(xs, anchor, Rx, out);
}